// GlobalQueryAttentionPool_9680856285826
// MI455X (gfx1250) — compile-verified
//
#include <hip/hip_runtime.h>
#include <math.h>
#include <stdint.h>

#define HIDDEN   768
#define HEADS    8
#define HEAD_DIM 96
#define NB       16
#define SEQ      8192

typedef __attribute__((ext_vector_type(2))) float v2f;
typedef __attribute__((ext_vector_type(8))) float v8f;
typedef __attribute__((ext_vector_type(4))) unsigned int v4u;
typedef __attribute__((ext_vector_type(8))) int v8i;
typedef __attribute__((ext_vector_type(4))) int v4i;

#if defined(__gfx1250__) && __has_builtin(__builtin_amdgcn_tensor_load_to_lds)
#define HAVE_TDM 1
#else
#define HAVE_TDM 0
#endif

// D = A(16x4) * B(4x16) + C, fp32 WMMA (wave32)
__device__ __forceinline__ v8f wmma4(v2f a, v2f b, v8f c) {
  return __builtin_amdgcn_wmma_f32_16x16x4_f32(false, a, false, b, (short)0, c,
                                               false, false);
}

#if HAVE_TDM
// 2D TDM tile load: tile_rows x tile_cols fp32 elements, row stride
// row_stride_elems, global -> LDS (contiguous, row-major). D# per ISA ch.8.
// 6-arg builtin: (uint32x4 g0, int32x8 g1, int32x4, int32x4, int32x8, i32 cpol)
__device__ __forceinline__ void tdm_load_2d(unsigned lds_off,
                                            const float* gaddr,
                                            int tile_cols, int tile_rows,
                                            int row_stride_elems) {
  unsigned long long ga = (unsigned long long)(uintptr_t)gaddr;
  v4u g0;
  g0.x = 1u;                                   // count=1, no gather
  g0.y = lds_off;                              // LDS byte address
  g0.z = (unsigned)(ga & 0xffffffffu);         // global_addr[31:0]
  g0.w = (unsigned)((ga >> 32) & 0x1ffffffu)   // global_addr[56:32]
         | (2u << 30);                         // type = 2 ("image")
  v8i g1;
  g1[0] = (int)(2u << 16);                     // data_size = 4B
  g1[1] = (int)(((unsigned)row_stride_elems & 0xffffu) << 16);  // tensor_dim0 lo
  g1[2] = 0;                                   // tensor_dim0 hi / tensor_dim1 lo
  g1[3] = (int)(16u | ((unsigned)tile_cols << 16)); // dim1 hi (1<<20) | tile_dim0
  g1[4] = tile_rows;                           // tile_dim1 (tile_dim2 = 0)
  g1[5] = row_stride_elems;                    // tensor_dim0_stride lo32
  g1[6] = 0;
  g1[7] = 0;
  v4i gz4 = {0, 0, 0, 0};
  v8i gz8 = {0, 0, 0, 0, 0, 0, 0, 0};
  __builtin_amdgcn_tensor_load_to_lds(g0, g1, gz4, gz4, gz8, 0);
}
#endif

// ---------------------------------------------------------------------------
// K0: fold Wk @ q_h * scale into [768 x 16], stored pre-swizzled as WMMA
// B-tiles: wqp[s][lane][2], lane = n + 16*((k&3)>>1), j = k&1. Heads 8..15 = 0.
// ---------------------------------------------------------------------------
__global__ void k0_fold(const float* __restrict__ Wk,
                        const float* __restrict__ gq,
                        float* __restrict__ wqp) {
  int idx = blockIdx.x * blockDim.x + threadIdx.x;  // (k*16 + n)
  if (idx >= HIDDEN * 16) return;
  int k = idx >> 4, n = idx & 15;
  float v = 0.f;
  if (n < HEADS) {
    const float scale = 0.102062072615966f;  // 1/sqrt(96)
    const float* wrow = Wk + (size_t)k * HIDDEN + n * HEAD_DIM;
    const float* q = gq + n * HEAD_DIM;
    float acc = 0.f;
    for (int j = 0; j < HEAD_DIM; ++j) acc += wrow[j] * q[j];
    v = acc * scale;
  }
  int s = k >> 2;
  int lane = n + 16 * ((k & 3) >> 1);
  int j = k & 1;
  wqp[s * 64 + lane * 2 + j] = v;
}

// ---------------------------------------------------------------------------
// K1: scores[b][h][l] = x[b,l,:] . wq[:,h]  via WMMA 16x16x4, with x tiles
// staged into LDS by the Tensor Data Mover (double buffered, wave 0 issues).
// Block = 128 rows of x (8 waves x 16-row tiles); 12 chunks of 64 columns.
// Scores land in the zero-padded A-operand buffer w16[b][head][l].
// ---------------------------------------------------------------------------
#define K1_ROWS   128
#define K1_CHUNK  64
#define K1_NCHUNK 12

__global__ __launch_bounds__(256) void k1_scores(const float* __restrict__ x,
                                                 const float* __restrict__ wqp,
                                                 float* __restrict__ w16) {
  extern __shared__ float smem[];
  float* lwq = smem;                     // 192*64 = 12288 floats (48 KB)
  float* xb0 = smem + 12288;             // 128*64 =  8192 floats (32 KB)
  float* xb1 = xb0 + K1_ROWS * K1_CHUNK; // 32 KB

  for (int i = threadIdx.x; i < 192 * 64; i += 256) lwq[i] = wqp[i];

  int lane = threadIdx.x & 31;
  int wave = threadIdx.x >> 5;
  const float* xblk = x + (size_t)blockIdx.x * K1_ROWS * HIDDEN;
  bool issuer = (wave == 0);

#if HAVE_TDM
  if (issuer) {
    tdm_load_2d((unsigned)(uintptr_t)xb0, xblk + 0 * K1_CHUNK, K1_CHUNK,
                K1_ROWS, HIDDEN);
    tdm_load_2d((unsigned)(uintptr_t)xb1, xblk + 1 * K1_CHUNK, K1_CHUNK,
                K1_ROWS, HIDDEN);
  }
#endif

  int m = lane & 15;
  int kh = (lane >> 4) << 1;  // 0 or 2
  v8f c = {};
  for (int ch = 0; ch < K1_NCHUNK; ++ch) {
    float* buf = (ch & 1) ? xb1 : xb0;
#if HAVE_TDM
    if (issuer) {
      if (ch < K1_NCHUNK - 1) __builtin_amdgcn_s_wait_tensorcnt(1);
      else                    __builtin_amdgcn_s_wait_tensorcnt(0);
    }
    __syncthreads();  // chunk ch visible (also covers lwq staging on ch==0)
#else
    __syncthreads();
    for (int i = threadIdx.x; i < K1_ROWS * K1_CHUNK; i += 256) {
      int r = i / K1_CHUNK, cc = i % K1_CHUNK;
      buf[i] = xblk[(size_t)r * HIDDEN + ch * K1_CHUNK + cc];
    }
    __syncthreads();
#endif
    const float* ar = buf + (size_t)(wave * 16 + m) * K1_CHUNK + kh;
    int sbase = ch * (K1_CHUNK / 4);
#pragma unroll 4
    for (int s = 0; s < K1_CHUNK / 4; ++s) {
      v2f a = *(const v2f*)(ar + 4 * s);
      v2f b = *(const v2f*)(lwq + (sbase + s) * 64 + lane * 2);
      c = wmma4(a, b, c);
    }
    __syncthreads();  // everyone done reading buf before it is overwritten
#if HAVE_TDM
    if (issuer && ch + 2 < K1_NCHUNK) {
      tdm_load_2d((unsigned)(uintptr_t)buf, xblk + (ch + 2) * K1_CHUNK,
                  K1_CHUNK, K1_ROWS, HIDDEN);
    }
#endif
  }

  // C layout: VGPR r -> row M = r + 8*(lane>>4), col N = lane&15 (= head)
  int tile = blockIdx.x * 8 + wave;  // 16-row tile index, 512 per batch
  int n = lane & 15;
  if (n < HEADS) {
    int bidx = tile >> 9;
    int posb = (tile & 511) * 16 + 8 * (lane >> 4);
    float* out = w16 + (size_t)(bidx * 16 + n) * SEQ;
#pragma unroll
    for (int r = 0; r < 8; ++r) out[posb + r] = c[r];
  }
}

// ---------------------------------------------------------------------------
// K2: block (b,m): m<8 -> in-place softmax over L; m>=8 -> zero padding row.
// ---------------------------------------------------------------------------
__global__ __launch_bounds__(256) void k2_softmax(float* __restrict__ w16) {
  __shared__ float red[256];
  int bm = blockIdx.x;  // 0..255
  float* p = w16 + (size_t)bm * SEQ;
  int t = threadIdx.x;
  if ((bm & 15) >= HEADS) {
    for (int i = t; i < SEQ; i += 256) p[i] = 0.f;
    return;
  }
  float mx = -1e30f;
  for (int i = t; i < SEQ; i += 256) mx = fmaxf(mx, p[i]);
  red[t] = mx;
  __syncthreads();
  for (int s = 128; s > 0; s >>= 1) {
    if (t < s) red[t] = fmaxf(red[t], red[t + s]);
    __syncthreads();
  }
  float m = red[0];
  __syncthreads();
  float sum = 0.f;
  for (int i = t; i < SEQ; i += 256) sum += __expf(p[i] - m);
  red[t] = sum;
  __syncthreads();
  for (int s = 128; s > 0; s >>= 1) {
    if (t < s) red[t] += red[t + s];
    __syncthreads();
  }
  float inv = 1.0f / red[0];
  for (int i = t; i < SEQ; i += 256) p[i] = __expf(p[i] - m) * inv;
}

// ---------------------------------------------------------------------------
// K3: pooled-x partials via WMMA: C[16x64] += W16[16x4] * X[4x64] over a
// 1024-row K-chunk. Block = (b, 64-col chunk), wave = K-chunk (8 waves).
// Each wave TDM-stages its own 32x64 x tiles into a private LDS double
// buffer (per-wave TENSORcnt, no barriers). Heads 8..15 are zero padding.
// ---------------------------------------------------------------------------
__global__ __launch_bounds__(256) void k3_pool(const float* __restrict__ x,
                                               const float* __restrict__ w16,
                                               float* __restrict__ pxp) {
  extern __shared__ float smem[];  // 8 waves * 2 * (32*64) floats = 128 KB
  int b = blockIdx.x / 12;
  int nc = blockIdx.x % 12;
  int lane = threadIdx.x & 31;
  int kc = __builtin_amdgcn_readfirstlane((int)(threadIdx.x >> 5));  // 0..7
  int m = lane & 15;
  int kh = (lane >> 4) << 1;
  int n = lane & 15;

  const float* ap = w16 + (size_t)(b * 16 + m) * SEQ + kc * 1024 + kh;

  v8f c0 = {}, c1 = {}, c2 = {}, c3 = {};

#if HAVE_TDM
  float* lbuf = smem + kc * 4096;  // two 2048-float (8 KB) buffers per wave
  const float* xtile0 =
      x + ((size_t)b * SEQ + (size_t)kc * 1024) * HIDDEN + nc * 64;

  tdm_load_2d((unsigned)(uintptr_t)lbuf, xtile0, 64, 32, HIDDEN);
  tdm_load_2d((unsigned)(uintptr_t)(lbuf + 2048), xtile0 + 32 * HIDDEN, 64, 32,
              HIDDEN);

  for (int ch = 0; ch < 32; ++ch) {
    if (ch < 31) __builtin_amdgcn_s_wait_tensorcnt(1);
    else         __builtin_amdgcn_s_wait_tensorcnt(0);
    const float* bb = lbuf + (ch & 1) * 2048;
    const float* apc = ap + ch * 32;
#pragma unroll
    for (int sl = 0; sl < 8; ++sl) {
      v2f a = *(const v2f*)(apc + 4 * sl);
      const float* br = bb + (sl * 4 + kh) * 64 + n;
      v2f b0, b1, b2, b3;
      b0.x = br[0];  b0.y = br[64];
      b1.x = br[16]; b1.y = br[80];
      b2.x = br[32]; b2.y = br[96];
      b3.x = br[48]; b3.y = br[112];
      c0 = wmma4(a, b0, c0);
      c1 = wmma4(a, b1, c1);
      c2 = wmma4(a, b2, c2);
      c3 = wmma4(a, b3, c3);
    }
    if (ch + 2 < 32) {  // overwrite the buffer just consumed
      tdm_load_2d((unsigned)(uintptr_t)(lbuf + (ch & 1) * 2048),
                  xtile0 + (size_t)(ch + 2) * 32 * HIDDEN, 64, 32, HIDDEN);
    }
  }
#else
  const float* xp =
      x + ((size_t)b * SEQ + (size_t)kc * 1024 + kh) * HIDDEN + nc * 64 + n;
#pragma unroll 2
  for (int s = 0; s < 256; ++s) {
    v2f a = *(const v2f*)(ap + 4 * s);
    const float* xr = xp + (size_t)(4 * s) * HIDDEN;
    v2f b0, b1, b2, b3;
    b0.x = xr[0];  b0.y = xr[HIDDEN];
    b1.x = xr[16]; b1.y = xr[HIDDEN + 16];
    b2.x = xr[32]; b2.y = xr[HIDDEN + 32];
    b3.x = xr[48]; b3.y = xr[HIDDEN + 48];
    c0 = wmma4(a, b0, c0);
    c1 = wmma4(a, b1, c1);
    c2 = wmma4(a, b2, c2);
    c3 = wmma4(a, b3, c3);
  }
#endif

  // C layout: row M = r + 8*(lane>>4) = head; col = lane&15. Heads >= 8 are 0.
  if (lane < 16) {
    float* o = pxp + (size_t)(kc * 16 + b) * 8 * HIDDEN + nc * 64 + n;
#pragma unroll
    for (int r = 0; r < 8; ++r) {
      o[(size_t)r * HIDDEN + 0] = c0[r];
      o[(size_t)r * HIDDEN + 16] = c1[r];
      o[(size_t)r * HIDDEN + 32] = c2[r];
      o[(size_t)r * HIDDEN + 48] = c3[r];
    }
  }
}

// ---------------------------------------------------------------------------
// K4: reduce the 8 K-chunk partials: px[e] = sum_kc pxp[kc][e]
// ---------------------------------------------------------------------------
__global__ void k4_reduce(const float* __restrict__ pxp, float* __restrict__ px) {
  int e = blockIdx.x * blockDim.x + threadIdx.x;
  if (e >= 98304) return;
  float s = 0.f;
#pragma unroll
  for (int lc = 0; lc < 8; ++lc) s += pxp[(size_t)lc * 98304 + e];
  px[e] = s;
}

// ---------------------------------------------------------------------------
// K5a: pooled[b, h*96+j] = px[b,h,:] @ Wv[:, h*96+j] + bv   (WMMA, M=16=batch)
// ---------------------------------------------------------------------------
__global__ __launch_bounds__(32) void k5a_ctx(const float* __restrict__ px,
                                              const float* __restrict__ Wv,
                                              const float* __restrict__ bv,
                                              float* __restrict__ pooled) {
  int h = blockIdx.x / 6, nt = blockIdx.x % 6;
  int lane = threadIdx.x;
  int m = lane & 15;          // batch index
  int kh = (lane >> 4) << 1;  // 0 or 2
  int col = h * HEAD_DIM + nt * 16 + (lane & 15);
  const float* ar = px + (size_t)(m * HEADS + h) * HIDDEN + kh;

  v8f c = {};
  for (int s = 0; s < 192; ++s) {
    int k = 4 * s + kh;
    v2f a = *(const v2f*)(ar + 4 * s);
    v2f b;
    b.x = Wv[(size_t)k * HIDDEN + col];
    b.y = Wv[(size_t)(k + 1) * HIDDEN + col];
    c = wmma4(a, b, c);
  }
  float bvv = bv[col];
#pragma unroll
  for (int r = 0; r < 8; ++r) {
    int bi = r + 8 * (lane >> 4);
    pooled[(size_t)bi * HIDDEN + col] = c[r] + bvv;
  }
}

// ---------------------------------------------------------------------------
// K5b: out = pooled @ Wo + bo   (WMMA, M=16=batch). One wave per n-tile.
// ---------------------------------------------------------------------------
__global__ __launch_bounds__(32) void k5b_out(const float* __restrict__ pooled,
                                              const float* __restrict__ Wo,
                                              const float* __restrict__ bo,
                                              float* __restrict__ out) {
  int nt = blockIdx.x;
  int lane = threadIdx.x;
  int m = lane & 15;
  int kh = (lane >> 4) << 1;
  int col = nt * 16 + (lane & 15);
  const float* ar = pooled + (size_t)m * HIDDEN + kh;

  v8f c = {};
  for (int s = 0; s < 192; ++s) {
    int k = 4 * s + kh;
    v2f a = *(const v2f*)(ar + 4 * s);
    v2f b;
    b.x = Wo[(size_t)k * HIDDEN + col];
    b.y = Wo[(size_t)(k + 1) * HIDDEN + col];
    c = wmma4(a, b, c);
  }
  float bov = bo[col];
#pragma unroll
  for (int r = 0; r < 8; ++r) {
    int bi = r + 8 * (lane >> 4);
    out[(size_t)bi * HIDDEN + col] = c[r] + bov;
  }
}

extern "C" void kernel_launch(void* const* d_in, const int* in_sizes, int n_in,
                              void* d_out, int out_size, void* d_ws,
                              size_t ws_size, hipStream_t stream) {
  (void)in_sizes; (void)n_in; (void)out_size; (void)ws_size;
  const float* x  = (const float*)d_in[0];
  const float* gq = (const float*)d_in[1];
  const float* Wk = (const float*)d_in[2];
  // d_in[3] = bk: per-(b,h) constant added to all scores -> cancels in softmax
  const float* Wv = (const float*)d_in[4];
  const float* bv = (const float*)d_in[5];
  const float* Wo = (const float*)d_in[6];
  const float* bo = (const float*)d_in[7];
  float* out = (float*)d_out;

  float* ws = (float*)d_ws;
  float* wqp    = ws;                 // 12288 floats (swizzled B-tiles)
  float* w16    = ws + 16384;         // 16*16*8192 = 2097152 floats
  float* pxp    = w16 + 2097152;      // 8*16*8*768 = 786432 floats
  float* px     = pxp + 786432;       // 98304 floats
  float* pooled = px + 98304;         // 12288 floats

  const size_t k1_lds = (12288 + 2 * K1_ROWS * K1_CHUNK) * sizeof(float);
  const size_t k3_lds = 8 * 2 * 32 * 64 * sizeof(float);  // 128 KB

  k0_fold   <<<48,   256, 0,      stream>>>(Wk, gq, wqp);
  k1_scores <<<1024, 256, k1_lds, stream>>>(x, wqp, w16);
  k2_softmax<<<256,  256, 0,      stream>>>(w16);
  k3_pool   <<<192,  256, k3_lds, stream>>>(x, w16, pxp);
  k4_reduce <<<384,  256, 0,      stream>>>(pxp, px);
  k5a_ctx   <<<48,    32, 0,      stream>>>(px, Wv, bv, pooled);
  k5b_out   <<<48,    32, 0,      stream>>>(pooled, Wo, bo, out);
}